// Model_lite_22033182228933
// MI455X (gfx1250) — compile-verified
//
#include <hip/hip_runtime.h>
#include <hip/hip_bf16.h>

// ---------------- constants ----------------
#define H        4096
#define NH       32
#define HD       128
#define S_PAST   8192
#define NK       8193            // S+1
#define KSTRIDE  4096            // floats per key row (NH*HD)
#define INNER    16384
#define SCST     8224            // padded scores row stride (zero pad [NK,SCST))
#define QK_COEFF 1.0f            // layer 0: (LAYER_ID+1)
#define INV_COEFF 0.08838834764831845f  // 1/(sqrt(128)*QK_COEFF)
#define ALPHA_C  7.483314773547883f     // sqrt(2*28)

typedef __attribute__((ext_vector_type(2))) float v2f;
typedef __attribute__((ext_vector_type(8))) float v8f;

// workspace float offsets
#define WS_RES1   0
#define WS_QKV    (WS_RES1 + H)            // 12288
#define WS_QS     (WS_QKV + 3*H)           // scaled rotated q [32][128]
#define WS_HID    (WS_QS + H)
#define WS_RES2   (WS_HID + H)
#define WS_CTX    (WS_RES2 + H)
#define WS_INNER  (WS_CTX + H)
#define WS_SCORES (WS_INNER + INNER)       // [32][SCST]

__device__ __forceinline__ float rdlane(float v, int l) {
  return __int_as_float(__builtin_amdgcn_readlane(__float_as_int(v), l));
}

// ---------------- layernorm (one block, 1024 thr = 32 waves) ----------------
__global__ __launch_bounds__(1024) void ln_kernel(
    const float* __restrict__ x, const float* __restrict__ w,
    const float* __restrict__ b, float* __restrict__ y) {
  __shared__ float s1[32], s2[32];
  __shared__ float sMu, sRstd;
  float sum = 0.f, sq = 0.f;
  for (int i = threadIdx.x; i < H; i += blockDim.x) {
    float v = x[i]; sum += v; sq += v * v;
  }
  for (int o = 16; o; o >>= 1) { sum += __shfl_down(sum, o); sq += __shfl_down(sq, o); }
  int wid = threadIdx.x >> 5, lane = threadIdx.x & 31;
  if (lane == 0) { s1[wid] = sum; s2[wid] = sq; }
  __syncthreads();
  if (wid == 0) {
    sum = s1[lane]; sq = s2[lane];
    for (int o = 16; o; o >>= 1) { sum += __shfl_down(sum, o); sq += __shfl_down(sq, o); }
    if (lane == 0) {
      float mu = sum * (1.0f / H);
      float var = sq * (1.0f / H) - mu * mu;
      sMu = mu; sRstd = rsqrtf(var + 1e-5f);
    }
  }
  __syncthreads();
  float mu = sMu, rstd = sRstd;
  for (int i = threadIdx.x; i < H; i += blockDim.x)
    y[i] = (x[i] - mu) * rstd * w[i] + b[i];
}

// ---------------- GEMV: 16 rows/block, wave/row, x in LDS ----------------
// mode 0: y = Wx+b   mode 1: y = res*ALPHA + Wx+b   mode 2: y = gelu(Wx+b)
__global__ __launch_bounds__(512) void gemv_kernel(
    const float* __restrict__ W, const float* __restrict__ x,
    const float* __restrict__ bias, const float* __restrict__ res,
    float* __restrict__ y, int cols, int mode) {
  extern __shared__ float xs[];
  const int tid = threadIdx.x;
  const int nc4 = cols >> 2;
  const float4* x4 = (const float4*)x;
  float4* xs4 = (float4*)xs;
  for (int i = tid; i < nc4; i += blockDim.x) xs4[i] = x4[i];
  __syncthreads();

  const int wave = tid >> 5, lane = tid & 31;
  const int row = blockIdx.x * 16 + wave;
  const float4* W4 = (const float4*)(W + (size_t)row * cols);
  float acc = 0.f;
#pragma unroll 4
  for (int i = lane; i < nc4; i += 32) {
    float4 w = W4[i], xx = xs4[i];
    acc = fmaf(w.x, xx.x, acc);
    acc = fmaf(w.y, xx.y, acc);
    acc = fmaf(w.z, xx.z, acc);
    acc = fmaf(w.w, xx.w, acc);
  }
  for (int o = 16; o; o >>= 1) acc += __shfl_down(acc, o);
  if (lane == 0) {
    float v = acc + bias[row];
    if (mode == 1) v = res[row] * ALPHA_C + v;
    else if (mode == 2) {
      float x3 = v * v * v;
      v = 0.5f * v * (1.0f + tanhf(0.7978845608028654f * (v + 0.044715f * x3)));
    }
    y[row] = v;
  }
}

// ---------------- rope: rotate q/k, scale q, emit new k/v row, zero ctx ----------------
__global__ __launch_bounds__(256) void rope_kernel(
    const float* __restrict__ qkv, const float* __restrict__ cosc,
    const float* __restrict__ sinc, const int* __restrict__ pos,
    const int* __restrict__ bpos, float* __restrict__ q_s,
    float* __restrict__ outK, float* __restrict__ outV,
    float* __restrict__ ctx) {
  int idx = blockIdx.x * blockDim.x + threadIdx.x;   // 0..4095
  int h = idx >> 7, d = idx & 127;
  int half = d >> 6, j = d & 63;
  int p = half ? bpos[0] : pos[0];
  float c = cosc[p * 64 + j], sn = sinc[p * 64 + j];
  int jo = (j < 32) ? (j + 32) : (j - 32);
  float sgn = (j < 32) ? -1.f : 1.f;

  int base = h * 384;
  float q  = qkv[base + d];
  float k  = qkv[base + 128 + d];
  float v  = qkv[base + 256 + d];
  float qo = qkv[base + (half << 6) + jo];
  float ko = qkv[base + 128 + (half << 6) + jo];

  float qr = q * c + sgn * qo * sn;
  float kr = k * c + sgn * ko * sn;

  q_s[idx] = qr * INV_COEFF;
  outK[(size_t)S_PAST * KSTRIDE + idx] = kr;
  outV[(size_t)S_PAST * KSTRIDE + idx] = v;
  ctx[idx] = 0.f;   // zero for the atomic-accumulate pass (stream-ordered)
}

// ---------------- attention scores via V_WMMA_F32_16X16X4_F32 ----------------
// Per wave: 16 keys (A rows), q chunk in B column 0, K=128 consumed in 32 steps.
// A rows are CLAMPED (no exec-mask branches); invalid D rows are never written.
// B broadcast uses v_readlane (uniform index) instead of ds_bpermute.
__global__ __launch_bounds__(256) void attn_scores_wmma(
    const float* __restrict__ Kc, const float* __restrict__ q_s,
    const unsigned char* __restrict__ mask, float* __restrict__ scores) {
  const int h = blockIdx.y;
  const int wave = threadIdx.x >> 5;
  const int lane = threadIdx.x & 31;
  const int kb = (blockIdx.x * 8 + wave) * 16;

  // lane L holds q_s[h][4L..4L+3]
  const float4 qv = ((const float4*)(q_s + h * 128))[lane];

  const int row = min(kb + (lane & 15), NK - 1);   // clamp: no branches in loop
  const float* abase = Kc + (size_t)row * KSTRIDE + h * 128 + ((lane >> 4) << 1);

  const bool sel = (lane & 15) == 0;   // lanes 0 and 16 carry B column 0
  const bool hi  = lane >= 16;

  v8f c = {0.f, 0.f, 0.f, 0.f, 0.f, 0.f, 0.f, 0.f};
#pragma unroll
  for (int i = 0; i < 32; ++i) {
    const float* p = abase + 4 * i;
    v2f a; a[0] = p[0]; a[1] = p[1];
    float q0 = rdlane(qv.x, i), q1 = rdlane(qv.y, i);
    float q2 = rdlane(qv.z, i), q3 = rdlane(qv.w, i);
    v2f b;
    b[0] = sel ? (hi ? q2 : q0) : 0.f;
    b[1] = sel ? (hi ? q3 : q1) : 0.f;
    c = __builtin_amdgcn_wmma_f32_16x16x4_f32(false, a, false, b, (short)0, c,
                                              false, false);
  }
  // D column 0: lane0 -> M=0..7 in c[0..7], lane16 -> M=8..15
  if (lane == 0 || lane == 16) {
    int off = (lane == 16) ? 8 : 0;
    for (int jj = 0; jj < 8; ++jj) {
      int key = kb + off + jj;
      if (key < NK) {
        float s = c[jj];
        s = mask[key] ? -10000.0f : s;
        scores[h * SCST + key] = s * QK_COEFF;
      }
    }
  }
}

// ---------------- softmax per head (block = head); zero-pads row to SCST ----------------
__global__ __launch_bounds__(256) void softmax_kernel(float* __restrict__ scores) {
  __shared__ float red[8];
  __shared__ float sM, sSum;
  float* s = scores + blockIdx.x * SCST;
  const int tid = threadIdx.x, wid = tid >> 5, lane = tid & 31;

  float m = -3.4e38f;
  for (int i = tid; i < NK; i += 256) m = fmaxf(m, s[i]);
  for (int o = 16; o; o >>= 1) m = fmaxf(m, __shfl_down(m, o));
  if (lane == 0) red[wid] = m;
  __syncthreads();
  if (tid == 0) {
    float mm = red[0];
    for (int i = 1; i < 8; ++i) mm = fmaxf(mm, red[i]);
    sM = mm;
  }
  __syncthreads();
  float M = sM, sum = 0.f;
  for (int i = tid; i < NK; i += 256) { float e = __expf(s[i] - M); s[i] = e; sum += e; }
  for (int o = 16; o; o >>= 1) sum += __shfl_down(sum, o);
  if (lane == 0) red[wid] = sum;
  __syncthreads();
  if (tid == 0) {
    float t = 0.f;
    for (int i = 0; i < 8; ++i) t += red[i];
    sSum = t;
  }
  __syncthreads();
  float inv = 1.0f / sSum;
  for (int i = tid; i < NK; i += 256) s[i] *= inv;
  for (int i = NK + tid; i < SCST; i += 256) s[i] = 0.f;  // pad for ctx pass
}

// ---------------- ctx = probs . V via WMMA (16 dims/wave, 8-way key split) ----------------
// Key split (1028) is a multiple of 4, so tail reads hit the zero-padded probs
// region -> no per-iteration conditionals, no cross-split double counting.
__global__ __launch_bounds__(256) void attn_ctx_wmma(
    const float* __restrict__ Vc, const float* __restrict__ probs,
    float* __restrict__ ctx) {
  const int h = blockIdx.x;
  const int split = blockIdx.y;          // 0..7
  const int wave = threadIdx.x >> 5;
  const int lane = threadIdx.x & 31;
  const int db = wave * 16;              // d-chunk for this wave

  const int kchunk = 1028;               // multiple of 4; 8*1028 >= NK
  const int k0 = split * kchunk;
  const int k1 = min(k0 + kchunk, NK);

  const int dd = lane & 15;
  const int koff = (lane >> 4) << 1;     // 0 (lanes 0-15) or 2 (lanes 16-31)
  const float* prow = probs + h * SCST;
  const float* vbase = Vc + h * 128 + db + dd;
  const bool sel = (lane & 15) == 0;

  v8f c = {0.f, 0.f, 0.f, 0.f, 0.f, 0.f, 0.f, 0.f};
#pragma unroll 4
  for (int kb = k0; kb < k1; kb += 4) {
    // A[m=dd][K]: lane<16 -> V rows kb,kb+1 ; lane>=16 -> rows kb+2,kb+3
    int r0 = min(kb + koff, NK - 1);     // clamp; garbage killed by b==0
    int r1 = min(kb + koff + 1, NK - 1);
    v2f a;
    a[0] = vbase[(size_t)r0 * KSTRIDE];
    a[1] = vbase[(size_t)r1 * KSTRIDE];
    // B column 0 = probs[kb..kb+3] (reads into zero pad are safe)
    v2f pv = *(const v2f*)(prow + kb + koff);
    v2f b;
    b[0] = sel ? pv[0] : 0.f;
    b[1] = sel ? pv[1] : 0.f;
    c = __builtin_amdgcn_wmma_f32_16x16x4_f32(false, a, false, b, (short)0, c,
                                              false, false);
  }
  if (lane == 0 || lane == 16) {
    int off = (lane == 16) ? 8 : 0;
    for (int jj = 0; jj < 8; ++jj)
      atomicAdd(&ctx[h * 128 + db + off + jj], c[jj]);
  }
}

// ---------------- launch ----------------
extern "C" void kernel_launch(void* const* d_in, const int* in_sizes, int n_in,
                              void* d_out, int out_size, void* d_ws, size_t ws_size,
                              hipStream_t stream) {
  const float* hidden = (const float*)d_in[0];
  const float* past_k = (const float*)d_in[1];
  const float* past_v = (const float*)d_in[2];
  const float* cosc   = (const float*)d_in[3];
  const float* sinc   = (const float*)d_in[4];
  const float* ln1w   = (const float*)d_in[5];
  const float* ln1b   = (const float*)d_in[6];
  const float* qkvw   = (const float*)d_in[7];
  const float* qkvb   = (const float*)d_in[8];
  const float* densew = (const float*)d_in[9];
  const float* denseb = (const float*)d_in[10];
  const float* ln2w   = (const float*)d_in[11];
  const float* ln2b   = (const float*)d_in[12];
  const float* fc1w   = (const float*)d_in[13];
  const float* fc1b   = (const float*)d_in[14];
  const float* fc2w   = (const float*)d_in[15];
  const float* fc2b   = (const float*)d_in[16];
  const int* pos      = (const int*)d_in[17];
  const int* bpos     = (const int*)d_in[18];
  const unsigned char* mask = (const unsigned char*)d_in[19];

  float* out  = (float*)d_out;
  float* outK = out + H;
  float* outV = outK + (size_t)NK * KSTRIDE;
  float* ws   = (float*)d_ws;

  // bulk KV cache pass-through (D2D, graph-capture safe)
  hipMemcpyAsync(outK, past_k, (size_t)S_PAST * KSTRIDE * sizeof(float),
                 hipMemcpyDeviceToDevice, stream);
  hipMemcpyAsync(outV, past_v, (size_t)S_PAST * KSTRIDE * sizeof(float),
                 hipMemcpyDeviceToDevice, stream);

  ln_kernel<<<1, 1024, 0, stream>>>(hidden, ln1w, ln1b, ws + WS_RES1);
  gemv_kernel<<<(3 * H) / 16, 512, H * sizeof(float), stream>>>(
      qkvw, ws + WS_RES1, qkvb, nullptr, ws + WS_QKV, H, 0);
  rope_kernel<<<16, 256, 0, stream>>>(ws + WS_QKV, cosc, sinc, pos, bpos,
                                      ws + WS_QS, outK, outV, ws + WS_CTX);
  attn_scores_wmma<<<dim3(65, 32), 256, 0, stream>>>(outK, ws + WS_QS, mask,
                                                     ws + WS_SCORES);
  softmax_kernel<<<32, 256, 0, stream>>>(ws + WS_SCORES);
  attn_ctx_wmma<<<dim3(32, 8), 256, 0, stream>>>(outV, ws + WS_SCORES,
                                                 ws + WS_CTX);
  gemv_kernel<<<H / 16, 512, H * sizeof(float), stream>>>(
      densew, ws + WS_CTX, denseb, ws + WS_RES1, ws + WS_HID, H, 1);
  ln_kernel<<<1, 1024, 0, stream>>>(ws + WS_HID, ln2w, ln2b, ws + WS_RES2);
  gemv_kernel<<<INNER / 16, 512, H * sizeof(float), stream>>>(
      fc1w, ws + WS_RES2, fc1b, nullptr, ws + WS_INNER, H, 2);
  gemv_kernel<<<H / 16, 512, INNER * sizeof(float), stream>>>(
      fc2w, ws + WS_INNER, fc2b, ws + WS_RES2, out, INNER, 1);
}